// AttributeDecoder_39591008534760
// MI455X (gfx1250) — compile-verified
//
#include <hip/hip_runtime.h>

typedef __attribute__((ext_vector_type(2))) float v2f;
typedef __attribute__((ext_vector_type(8))) float v8f;

// ---------------- degree / normalization ----------------

__global__ void k_init_deg(float* __restrict__ deg, int n) {
  int i = blockIdx.x * blockDim.x + threadIdx.x;
  if (i < n) deg[i] = 1.0f;  // self-loop contribution
}

__global__ void k_count_deg(const long long* __restrict__ dst, float* __restrict__ deg, int e) {
  int i = blockIdx.x * blockDim.x + threadIdx.x;
  if (i < e) atomicAdd(&deg[(size_t)dst[i]], 1.0f);
}

__global__ void k_dinv(float* __restrict__ deg, int n) {
  int i = blockIdx.x * blockDim.x + threadIdx.x;
  if (i < n) deg[i] = rsqrtf(deg[i]);  // deg >= 1 always (self-loops)
}

// ---------------- dense GEMM via V_WMMA_F32_16X16X4_F32 ----------------
// One wave computes one 16x16 tile of out = A[M,K] @ W[K,NOUT].
// fp32 WMMA fragment layouts per CDNA5 ISA 7.12.2:
//   A 16x4 : lanes 0-15 -> K = k,k+1 ; lanes 16-31 -> K = k+2,k+3 (row = lane&15)
//   B 4x16 : same K split across lane halves (col = lane&15)
//   C/D    : VGPR r -> row r (lanes 0-15) / row r+8 (lanes 16-31)
template <int K, int NOUT, bool RELU>
__global__ void k_gemm_wmma(const float* __restrict__ A, const float* __restrict__ W,
                            float* __restrict__ out, int M, int totalTiles) {
  const int wave = threadIdx.x >> 5;
  const int lane = threadIdx.x & 31;
  const int tile = blockIdx.x * (blockDim.x >> 5) + wave;
  if (tile >= totalTiles) return;  // wave-uniform: EXEC stays all-1s for WMMA

  constexpr int NT = NOUT / 16;
  const int tm = tile / NT;
  const int tn = tile % NT;

  const int row = lane & 15;       // M within tile (A) / N within tile (B, C/D)
  const int hi  = lane >> 4;       // K-half selector

  const float* __restrict__ arow = A + (size_t)(tm * 16 + row) * K;
  const float* __restrict__ wcol = W + (size_t)tn * 16 + row;

  v8f c = {};
#pragma unroll
  for (int k = 0; k < K; k += 4) {
    const int ka = k + 2 * hi;
    float a0 = arow[ka];
    float a1 = arow[ka + 1];
    if (RELU) { a0 = fmaxf(a0, 0.0f); a1 = fmaxf(a1, 0.0f); }
    v2f a = {a0, a1};
    v2f b = {wcol[(size_t)ka * NOUT], wcol[(size_t)(ka + 1) * NOUT]};
    c = __builtin_amdgcn_wmma_f32_16x16x4_f32(false, a, false, b, (short)0, c, false, false);
  }

  float* __restrict__ obase = out + (size_t)(tm * 16 + 8 * hi) * NOUT + (size_t)tn * 16 + row;
#pragma unroll
  for (int r = 0; r < 8; r++) obase[(size_t)r * NOUT] = c[r];
}

// ---------------- self-loop + bias init: out[i,:] = b + h[i,:]*dinv[i]^2 ----------------
template <int F>
__global__ void k_selfloop_bias(const float* __restrict__ h, const float* __restrict__ dinv,
                                const float* __restrict__ bias, float* __restrict__ out, int n) {
  constexpr int Q = F / 4;
  int t = blockIdx.x * blockDim.x + threadIdx.x;
  if (t >= n * Q) return;
  int i = t / Q;
  int f = t % Q;
  float di = dinv[i];
  float s = di * di;
  float4 hv = ((const float4*)h)[(size_t)i * Q + f];
  float4 bv = ((const float4*)bias)[f];
  float4 o;
  o.x = bv.x + hv.x * s;
  o.y = bv.y + hv.y * s;
  o.z = bv.z + hv.z * s;
  o.w = bv.w + hv.w * s;
  ((float4*)out)[(size_t)i * Q + f] = o;
}

// ---------------- edge scatter: out[d,:] += h[s,:] * dinv[s]*dinv[d] ----------------
// One wave32 per edge; lane covers F/32 consecutive floats -> fully coalesced row read.
template <int F>
__global__ void k_edge_scatter(const float* __restrict__ h, const long long* __restrict__ src,
                               const long long* __restrict__ dst, const float* __restrict__ dinv,
                               float* __restrict__ out, int e) {
  constexpr int FPL = F / 32;
  int t = blockIdx.x * blockDim.x + threadIdx.x;
  int edge = t >> 5;
  int lane = t & 31;
  if (edge >= e) return;
  long long s = src[edge];
  long long d = dst[edge];
  float norm = dinv[s] * dinv[d];
  const float* __restrict__ hs = h + (size_t)s * F + lane * FPL;
  float* __restrict__ od = out + (size_t)d * F + lane * FPL;
  if (FPL == 4) {
    float4 v = *(const float4*)hs;
    atomicAdd(&od[0], v.x * norm);
    atomicAdd(&od[1], v.y * norm);
    atomicAdd(&od[2], v.z * norm);
    atomicAdd(&od[3], v.w * norm);
  } else {
    float2 v = *(const float2*)hs;
    atomicAdd(&od[0], v.x * norm);
    atomicAdd(&od[1], v.y * norm);
  }
}

extern "C" void kernel_launch(void* const* d_in, const int* in_sizes, int n_in,
                              void* d_out, int out_size, void* d_ws, size_t ws_size,
                              hipStream_t stream) {
  const float* z        = (const float*)d_in[0];
  const long long* ei   = (const long long*)d_in[1];
  const float* W1       = (const float*)d_in[2];
  const float* b1       = (const float*)d_in[3];
  const float* W2       = (const float*)d_in[4];
  const float* b2       = (const float*)d_in[5];

  const int n = in_sizes[0] / 64;   // 100000
  const int e = in_sizes[1] / 2;    // 1600000
  const long long* src = ei;
  const long long* dst = ei + e;

  // workspace layout (floats): dinv[n] | h1[n*128] | out1[n*128] | h2[n*64]
  float* ws   = (float*)d_ws;
  size_t off  = ((size_t)n + 255) & ~(size_t)255;
  float* dinv = ws;
  float* h1   = ws + off;
  float* out1 = h1 + (size_t)n * 128;
  float* h2   = out1 + (size_t)n * 128;
  float* out2 = (float*)d_out;

  const int B = 256;

  // degree -> dinv
  k_init_deg<<<(n + B - 1) / B, B, 0, stream>>>(dinv, n);
  k_count_deg<<<(e + B - 1) / B, B, 0, stream>>>(dst, dinv, e);
  k_dinv<<<(n + B - 1) / B, B, 0, stream>>>(dinv, n);

  const int tilesM = (n + 15) / 16;  // 6250

  // ---- layer 1: h1 = z @ W1 ; out1 = scatter(norm * h1) + b1 ----
  {
    int tiles = tilesM * (128 / 16);
    k_gemm_wmma<64, 128, false><<<(tiles + 3) / 4, 128, 0, stream>>>(z, W1, h1, n, tiles);
  }
  {
    long long t = (long long)n * 32;  // n * F/4
    k_selfloop_bias<128><<<(int)((t + B - 1) / B), B, 0, stream>>>(h1, dinv, b1, out1, n);
  }
  {
    long long t = (long long)e * 32;
    k_edge_scatter<128><<<(int)((t + B - 1) / B), B, 0, stream>>>(h1, src, dst, dinv, out1, e);
  }

  // ---- layer 2: h2 = relu(out1) @ W2 ; out2 = scatter(norm * h2) + b2 ----
  {
    int tiles = tilesM * (64 / 16);
    k_gemm_wmma<128, 64, true><<<(tiles + 3) / 4, 128, 0, stream>>>(out1, W2, h2, n, tiles);
  }
  {
    long long t = (long long)n * 16;  // n * F/4
    k_selfloop_bias<64><<<(int)((t + B - 1) / B), B, 0, stream>>>(h2, dinv, b2, out2, n);
  }
  {
    long long t = (long long)e * 32;
    k_edge_scatter<64><<<(int)((t + B - 1) / B), B, 0, stream>>>(h2, src, dst, dinv, out2, e);
  }
}